// ProcessingBlock_15040975470802
// MI455X (gfx1250) — compile-verified
//
#include <hip/hip_runtime.h>
#include <math.h>

#define N_NODES 50000
#define N_EDGES 800000
#define C_DIM   128
#define ED_DIM  16
#define NEG_SLOPE 0.2f

typedef __attribute__((ext_vector_type(2))) float v2f;
typedef __attribute__((ext_vector_type(8))) float v8f;

__device__ __forceinline__ float lrelu(float x) {
    return x > 0.0f ? x : NEG_SLOPE * x;
}

// float atomic max via signed/unsigned integer atomics (valid with -inf init)
__device__ __forceinline__ void atomicMaxF(float* addr, float val) {
    if (val >= 0.0f) atomicMax((int*)addr, __float_as_int(val));
    else             atomicMin((unsigned int*)addr, __float_as_uint(val));
}

// ---------------------------------------------------------------------------
// K0: init per-node accumulators
__global__ __launch_bounds__(256)
void init_kernel(float* __restrict__ sumR, float* __restrict__ sumI,
                 float* __restrict__ cnt,  float* __restrict__ mR, float* __restrict__ mI) {
    int n = blockIdx.x * blockDim.x + threadIdx.x;
    if (n < N_NODES) {
        sumR[n] = 0.0f; sumI[n] = 0.0f; cnt[n] = 0.0f;
        mR[n] = -INFINITY; mI[n] = -INFINITY;
    }
}

// ---------------------------------------------------------------------------
// K1: v = W_edge @ att_edge (16-vector) for both param sets. 1 block, 32 thr.
__global__ void vcalc_kernel(const float* __restrict__ WeR, const float* __restrict__ aeR,
                             const float* __restrict__ WeI, const float* __restrict__ aeI,
                             float* __restrict__ vR, float* __restrict__ vI) {
    int t = threadIdx.x;
    const float* W = (t < 16) ? WeR : WeI;
    const float* a = (t < 16) ? aeR : aeI;
    int r = t & 15;
    float s = 0.0f;
    for (int j = 0; j < C_DIM; ++j) s += W[r * C_DIM + j] * a[j];
    if (t < 16) vR[r] = s; else vI[r] = s;
}

// ---------------------------------------------------------------------------
// K2: WMMA GEMM  h(50000x128) @ [W_r | Wres_r | W_i | Wres_i](128x512)
// One wave per 16x64 output tile: 4 f32 WMMA accumulators share one A
// fragment per K-step (4x WMMA density per A-load; h re-read 8x instead of
// 32x). 3125 row-tiles x 8 col-tiles = 25000 waves. Each 64-wide tile lies
// inside a single weight matrix (64 | 128), so matrix select is wave-uniform.
// Residual columns get bias folded in (initializes the out accumulators).
__global__ __launch_bounds__(256)
void gemm_kernel(const float* __restrict__ h,
                 const float* __restrict__ Wr,   const float* __restrict__ WresR, const float* __restrict__ biasR,
                 const float* __restrict__ Wi,   const float* __restrict__ WresI, const float* __restrict__ biasI,
                 float* __restrict__ Xr, float* __restrict__ OaccR,
                 float* __restrict__ Xi, float* __restrict__ OaccI) {
    const int wave = (blockIdx.x * blockDim.x + threadIdx.x) >> 5;
    const int lane = threadIdx.x & 31;
    const int bm   = wave >> 3;            // 0..3124  (row tile)
    const int bn   = wave & 7;             // 0..7     (64-wide col tile over 512 cols)
    const int half = lane >> 4;            // lane group (K split per ISA A layout)
    const int l16  = lane & 15;

    const int mat    = bn >> 1;            // which weight matrix
    const int colBlk = (bn & 1) * 64;      // column base within its 128-wide output
    const float* W; float* Dst; const float* bias = nullptr;
    if      (mat == 0) { W = Wr;    Dst = Xr;    }
    else if (mat == 1) { W = WresR; Dst = OaccR; bias = biasR; }
    else if (mat == 2) { W = Wi;    Dst = Xi;    }
    else               { W = WresI; Dst = OaccI; bias = biasI; }

    const int row  = bm * 16 + l16;        // A row held by this lane (both halves: M = lane&15)
    const int col0 = colBlk + l16;         // first B/D column held by this lane
    const float* __restrict__ hrow = h + (size_t)row * C_DIM;

    v8f acc0 = {0.f,0.f,0.f,0.f,0.f,0.f,0.f,0.f};
    v8f acc1 = acc0, acc2 = acc0, acc3 = acc0;

    #pragma unroll 2
    for (int k0 = 0; k0 < C_DIM; k0 += 4) {
        const int ka = k0 + 2 * half;      // A/B VGPR0 holds K=ka, VGPR1 holds K=ka+1
        v2f a;
        a.x = hrow[ka];
        a.y = hrow[ka + 1];
        const float* w0 = W + (size_t)ka * C_DIM + col0;
        const float* w1 = W + (size_t)(ka + 1) * C_DIM + col0;
        v2f b0, b1, b2, b3;
        b0.x = w0[0];  b0.y = w1[0];
        b1.x = w0[16]; b1.y = w1[16];
        b2.x = w0[32]; b2.y = w1[32];
        b3.x = w0[48]; b3.y = w1[48];
        acc0 = __builtin_amdgcn_wmma_f32_16x16x4_f32(false, a, false, b0, (short)0, acc0, false, false);
        acc1 = __builtin_amdgcn_wmma_f32_16x16x4_f32(false, a, false, b1, (short)0, acc1, false, false);
        acc2 = __builtin_amdgcn_wmma_f32_16x16x4_f32(false, a, false, b2, (short)0, acc2, false, false);
        acc3 = __builtin_amdgcn_wmma_f32_16x16x4_f32(false, a, false, b3, (short)0, acc3, false, false);
    }

    float bb0 = 0.f, bb1 = 0.f, bb2 = 0.f, bb3 = 0.f;
    if (bias) { bb0 = bias[col0]; bb1 = bias[col0+16]; bb2 = bias[col0+32]; bb3 = bias[col0+48]; }
    #pragma unroll
    for (int j = 0; j < 8; ++j) {
        const int orow = bm * 16 + j + 8 * half;   // C/D: VGPR j -> M = j (+8 for hi lanes)
        float* drow = Dst + (size_t)orow * C_DIM + col0;
        drow[0]  = acc0[j] + bb0;
        drow[16] = acc1[j] + bb1;
        drow[32] = acc2[j] + bb2;
        drow[48] = acc3[j] + bb3;
    }
}

// ---------------------------------------------------------------------------
// K3: per-node attention scalars  a_src = x.att_src, a_dst = x.att_dst (both sets)
// One wave per node, float4 per lane, shfl_xor tree reduction.
__global__ __launch_bounds__(256)
void node_dots_kernel(const float* __restrict__ Xr, const float* __restrict__ Xi,
                      const float* __restrict__ attSrcR, const float* __restrict__ attDstR,
                      const float* __restrict__ attSrcI, const float* __restrict__ attDstI,
                      float* __restrict__ aSrcR, float* __restrict__ aDstR,
                      float* __restrict__ aSrcI, float* __restrict__ aDstI) {
    const int gid = blockIdx.x * blockDim.x + threadIdx.x;
    const int n = gid >> 5, lane = gid & 31;
    if (n >= N_NODES) return;
    const int c0 = lane * 4;
    const float4 xr = *(const float4*)(Xr + (size_t)n * C_DIM + c0);
    const float4 xi = *(const float4*)(Xi + (size_t)n * C_DIM + c0);
    const float4 sr = *(const float4*)(attSrcR + c0);
    const float4 dr = *(const float4*)(attDstR + c0);
    const float4 si = *(const float4*)(attSrcI + c0);
    const float4 di = *(const float4*)(attDstI + c0);
    float p0 = xr.x*sr.x + xr.y*sr.y + xr.z*sr.z + xr.w*sr.w;
    float p1 = xr.x*dr.x + xr.y*dr.y + xr.z*dr.z + xr.w*dr.w;
    float p2 = xi.x*si.x + xi.y*si.y + xi.z*si.z + xi.w*si.w;
    float p3 = xi.x*di.x + xi.y*di.y + xi.z*di.z + xi.w*di.w;
    #pragma unroll
    for (int off = 16; off > 0; off >>= 1) {
        p0 += __shfl_xor(p0, off, 32);
        p1 += __shfl_xor(p1, off, 32);
        p2 += __shfl_xor(p2, off, 32);
        p3 += __shfl_xor(p3, off, 32);
    }
    if (lane == 0) { aSrcR[n] = p0; aDstR[n] = p1; aSrcI[n] = p2; aDstI[n] = p3; }
}

// ---------------------------------------------------------------------------
// K4: edge pass 1 — e = ea.v, alpha = lrelu(a_src[s]+a_dst[d]+e),
// segment sums of e / count, segment max of alpha. float4 (b128) row loads.
__global__ __launch_bounds__(256)
void edge1_kernel(const int* __restrict__ eidx, const float* __restrict__ ea,
                  const float* __restrict__ vR, const float* __restrict__ vI,
                  const float* __restrict__ aSrcR, const float* __restrict__ aDstR,
                  const float* __restrict__ aSrcI, const float* __restrict__ aDstI,
                  float* __restrict__ alphaR, float* __restrict__ alphaI,
                  float* __restrict__ sumR, float* __restrict__ sumI, float* __restrict__ cnt,
                  float* __restrict__ mR, float* __restrict__ mI) {
    int e = blockIdx.x * blockDim.x + threadIdx.x;
    if (e >= N_EDGES) return;
    const float4* row4 = (const float4*)(ea + (size_t)e * ED_DIM);
    const float4* vR4  = (const float4*)vR;
    const float4* vI4  = (const float4*)vI;
    float er = 0.0f, eii = 0.0f;
    #pragma unroll
    for (int q = 0; q < 4; ++q) {
        const float4 x = row4[q];
        const float4 r = vR4[q];
        const float4 i = vI4[q];
        er  += x.x*r.x + x.y*r.y + x.z*r.z + x.w*r.w;
        eii += x.x*i.x + x.y*i.y + x.z*i.z + x.w*i.w;
    }
    int s = eidx[e], d = eidx[N_EDGES + e];
    float ar = lrelu(aSrcR[s] + aDstR[d] + er);
    float ai = lrelu(aSrcI[s] + aDstI[d] + eii);
    alphaR[e] = ar; alphaI[e] = ai;
    atomicAdd(&sumR[d], er);
    atomicAdd(&sumI[d], eii);
    atomicAdd(&cnt[d], 1.0f);
    atomicMaxF(&mR[d], ar);
    atomicMaxF(&mI[d], ai);
}

// ---------------------------------------------------------------------------
// K5: per-node self-loop alpha (mean edge attr), finalize max, seed denom.
__global__ __launch_bounds__(256)
void node2_kernel(const float* __restrict__ aSrcR, const float* __restrict__ aDstR,
                  const float* __restrict__ aSrcI, const float* __restrict__ aDstI,
                  const float* __restrict__ sumR, const float* __restrict__ sumI,
                  const float* __restrict__ cnt,
                  float* __restrict__ mR, float* __restrict__ mI,
                  float* __restrict__ denR, float* __restrict__ denI,
                  float* __restrict__ eSelfR, float* __restrict__ eSelfI) {
    int n = blockIdx.x * blockDim.x + threadIdx.x;
    if (n >= N_NODES) return;
    float c = fmaxf(cnt[n], 1.0f);
    float ar = lrelu(aSrcR[n] + aDstR[n] + sumR[n] / c);
    float ai = lrelu(aSrcI[n] + aDstI[n] + sumI[n] / c);
    float mr = fmaxf(mR[n], ar);
    float mi = fmaxf(mI[n], ai);
    mR[n] = mr; mI[n] = mi;
    float esr = expf(ar - mr);
    float esi = expf(ai - mi);
    eSelfR[n] = esr; denR[n] = esr;
    eSelfI[n] = esi; denI[n] = esi;
}

// ---------------------------------------------------------------------------
// K6: edge pass 2 — ex = exp(alpha - m[d]) (stored in place), denom += ex.
__global__ __launch_bounds__(256)
void edge2_kernel(const int* __restrict__ eidx,
                  float* __restrict__ alphaR, float* __restrict__ alphaI,
                  const float* __restrict__ mR, const float* __restrict__ mI,
                  float* __restrict__ denR, float* __restrict__ denI) {
    int e = blockIdx.x * blockDim.x + threadIdx.x;
    if (e >= N_EDGES) return;
    int d = eidx[N_EDGES + e];
    float exr = expf(alphaR[e] - mR[d]);
    float exi = expf(alphaI[e] - mI[d]);
    alphaR[e] = exr; alphaI[e] = exi;
    atomicAdd(&denR[d], exr);
    atomicAdd(&denI[d], exi);
}

// ---------------------------------------------------------------------------
// K7: edge scatter — out[d] += (ex/denom[d]) * x[s]. One wave32 per edge,
// 4 channels per lane, both param sets. Accumulators are L2-resident.
__global__ __launch_bounds__(256)
void edge3_kernel(const int* __restrict__ eidx,
                  const float* __restrict__ exR, const float* __restrict__ exI,
                  const float* __restrict__ denR, const float* __restrict__ denI,
                  const float* __restrict__ Xr, const float* __restrict__ Xi,
                  float* __restrict__ OaccR, float* __restrict__ OaccI) {
    const int gid = blockIdx.x * blockDim.x + threadIdx.x;
    const int e = gid >> 5, lane = gid & 31;
    if (e >= N_EDGES) return;
    int s = eidx[e], d = eidx[N_EDGES + e];
    float wr = exR[e] / denR[d];
    float wi = exI[e] / denI[d];
    const int c0 = lane * 4;
    const float4 xr = *(const float4*)(Xr + (size_t)s * C_DIM + c0);
    const float4 xi = *(const float4*)(Xi + (size_t)s * C_DIM + c0);
    float* orp = OaccR + (size_t)d * C_DIM + c0;
    float* oip = OaccI + (size_t)d * C_DIM + c0;
    atomicAdd(orp + 0, wr * xr.x);
    atomicAdd(orp + 1, wr * xr.y);
    atomicAdd(orp + 2, wr * xr.z);
    atomicAdd(orp + 3, wr * xr.w);
    atomicAdd(oip + 0, wi * xi.x);
    atomicAdd(oip + 1, wi * xi.y);
    atomicAdd(oip + 2, wi * xi.z);
    atomicAdd(oip + 3, wi * xi.w);
}

// ---------------------------------------------------------------------------
// K8: self-loop contribution + select(reducible) + relu. One wave per node.
__global__ __launch_bounds__(256)
void node3_kernel(const float* __restrict__ Xr, const float* __restrict__ Xi,
                  const float* __restrict__ OaccR, float* __restrict__ OaccI_out,
                  const float* __restrict__ eSelfR, const float* __restrict__ eSelfI,
                  const float* __restrict__ denR, const float* __restrict__ denI,
                  const unsigned char* __restrict__ reducible) {
    const int gid = blockIdx.x * blockDim.x + threadIdx.x;
    const int n = gid >> 5, lane = gid & 31;
    if (n >= N_NODES) return;
    const int c0 = lane * 4;
    float wr = eSelfR[n] / denR[n];
    float wi = eSelfI[n] / denI[n];
    const float4 xr = *(const float4*)(Xr + (size_t)n * C_DIM + c0);
    const float4 xi = *(const float4*)(Xi + (size_t)n * C_DIM + c0);
    const float4 orv = *(const float4*)(OaccR     + (size_t)n * C_DIM + c0);
    const float4 oiv = *(const float4*)(OaccI_out + (size_t)n * C_DIM + c0);
    float4 vr, vi;
    vr.x = orv.x + wr * xr.x; vr.y = orv.y + wr * xr.y;
    vr.z = orv.z + wr * xr.z; vr.w = orv.w + wr * xr.w;
    vi.x = oiv.x + wi * xi.x; vi.y = oiv.y + wi * xi.y;
    vi.z = oiv.z + wi * xi.z; vi.w = oiv.w + wi * xi.w;
    bool red = reducible[n] != 0;
    float4 out;
    out.x = fmaxf(red ? vr.x : vi.x, 0.0f);
    out.y = fmaxf(red ? vr.y : vi.y, 0.0f);
    out.z = fmaxf(red ? vr.z : vi.z, 0.0f);
    out.w = fmaxf(red ? vr.w : vi.w, 0.0f);
    *(float4*)(OaccI_out + (size_t)n * C_DIM + c0) = out;
}

// ---------------------------------------------------------------------------
extern "C" void kernel_launch(void* const* d_in, const int* in_sizes, int n_in,
                              void* d_out, int out_size, void* d_ws, size_t ws_size,
                              hipStream_t stream) {
    (void)in_sizes; (void)n_in; (void)out_size; (void)ws_size;
    // inputs (setup_inputs order)
    const float*         h         = (const float*)d_in[0];
    const int*           edge_idx  = (const int*)d_in[1];
    const float*         edge_attr = (const float*)d_in[2];
    const unsigned char* reducible = (const unsigned char*)d_in[3];
    const float* red_W       = (const float*)d_in[4];
    const float* red_att_src = (const float*)d_in[5];
    const float* red_att_dst = (const float*)d_in[6];
    const float* red_W_edge  = (const float*)d_in[7];
    const float* red_att_edg = (const float*)d_in[8];
    const float* red_W_res   = (const float*)d_in[9];
    const float* red_bias    = (const float*)d_in[10];
    const float* irr_W       = (const float*)d_in[11];
    const float* irr_att_src = (const float*)d_in[12];
    const float* irr_att_dst = (const float*)d_in[13];
    const float* irr_W_edge  = (const float*)d_in[14];
    const float* irr_att_edg = (const float*)d_in[15];
    const float* irr_W_res   = (const float*)d_in[16];
    const float* irr_bias    = (const float*)d_in[17];

    float* out = (float*)d_out;                 // also serves as irr accumulator

    // workspace carve-up
    float* w = (float*)d_ws;
    const size_t NC = (size_t)N_NODES * C_DIM;
    float* Xr     = w; w += NC;
    float* Xi     = w; w += NC;
    float* OaccR  = w; w += NC;
    float* aSrcR  = w; w += N_NODES;
    float* aDstR  = w; w += N_NODES;
    float* aSrcI  = w; w += N_NODES;
    float* aDstI  = w; w += N_NODES;
    float* sumR   = w; w += N_NODES;
    float* sumI   = w; w += N_NODES;
    float* cnt    = w; w += N_NODES;
    float* mR     = w; w += N_NODES;
    float* mI     = w; w += N_NODES;
    float* denR   = w; w += N_NODES;
    float* denI   = w; w += N_NODES;
    float* eSelfR = w; w += N_NODES;
    float* eSelfI = w; w += N_NODES;
    float* alphaR = w; w += N_EDGES;
    float* alphaI = w; w += N_EDGES;
    float* vR     = w; w += 16;
    float* vI     = w; w += 16;

    const int TPB = 256;
    const int nodeBlocks     = (N_NODES + TPB - 1) / TPB;       // 196
    const int nodeWaveBlocks = (N_NODES * 32) / TPB;            // 6250 (exact)
    const int edgeBlocks     = (N_EDGES + TPB - 1) / TPB;       // 3125 (exact)
    const int edgeWaveBlocks = (N_EDGES * 32) / TPB;            // 100000 (exact)
    const int gemmBlocks     = (3125 * 8) / (TPB / 32);         // 3125 (exact)

    init_kernel<<<nodeBlocks, TPB, 0, stream>>>(sumR, sumI, cnt, mR, mI);
    vcalc_kernel<<<1, 32, 0, stream>>>(red_W_edge, red_att_edg, irr_W_edge, irr_att_edg, vR, vI);
    gemm_kernel<<<gemmBlocks, TPB, 0, stream>>>(h,
        red_W, red_W_res, red_bias, irr_W, irr_W_res, irr_bias,
        Xr, OaccR, Xi, out);
    node_dots_kernel<<<nodeWaveBlocks, TPB, 0, stream>>>(Xr, Xi,
        red_att_src, red_att_dst, irr_att_src, irr_att_dst,
        aSrcR, aDstR, aSrcI, aDstI);
    edge1_kernel<<<edgeBlocks, TPB, 0, stream>>>(edge_idx, edge_attr, vR, vI,
        aSrcR, aDstR, aSrcI, aDstI, alphaR, alphaI, sumR, sumI, cnt, mR, mI);
    node2_kernel<<<nodeBlocks, TPB, 0, stream>>>(aSrcR, aDstR, aSrcI, aDstI,
        sumR, sumI, cnt, mR, mI, denR, denI, eSelfR, eSelfI);
    edge2_kernel<<<edgeBlocks, TPB, 0, stream>>>(edge_idx, alphaR, alphaI, mR, mI, denR, denI);
    edge3_kernel<<<edgeWaveBlocks, TPB, 0, stream>>>(edge_idx, alphaR, alphaI,
        denR, denI, Xr, Xi, OaccR, out);
    node3_kernel<<<nodeWaveBlocks, TPB, 0, stream>>>(Xr, Xi, OaccR, out,
        eSelfR, eSelfI, denR, denI, reducible);
}